// graphcl_57329223467793
// MI455X (gfx1250) — compile-verified
//
#include <hip/hip_runtime.h>
#include <hip/hip_bf16.h>

typedef float v2f __attribute__((ext_vector_type(2)));
typedef float v4f __attribute__((ext_vector_type(4)));
typedef float v8f __attribute__((ext_vector_type(8)));

#define D 128

// ---------------------------------------------------------------------------
// Kernel 1: segment starts via binary search (batch is sorted).
// start[g] = first index i with batch[i] >= g
// ---------------------------------------------------------------------------
__global__ void seg_starts_kernel(const int* __restrict__ batch,
                                  int* __restrict__ start, int N, int G) {
    int g = blockIdx.x * blockDim.x + threadIdx.x;
    if (g >= G) return;
    int lo = 0, hi = N;
    while (lo < hi) {
        int mid = (lo + hi) >> 1;
        if (batch[mid] < g) lo = mid + 1; else hi = mid;
    }
    start[g] = lo;
}

// ---------------------------------------------------------------------------
// Kernel 2: per-graph weighted mean pooling. One 128-thread block per graph.
// Vectorized: thread t = rs*32 + tq; thread owns features [4*tq, 4*tq+4) and
// row-slot rs (nodes i with i%4==rs within each chunk). Each x access is a
// full 128-bit global_load_b128; 32 lanes cover one 512 B row. 4 row-slots
// process 4 nodes per step -> 4x fewer VMEM instructions on the 512 MB stream.
// Pass 1: segment max of imp (LDS tree reduce).
// Pass 2: chunked weights in LDS, acc4 += x4[n] * w[n]; final 4-way LDS reduce.
// ---------------------------------------------------------------------------
__global__ __launch_bounds__(128) void pool_kernel(
    const float* __restrict__ x, const float* __restrict__ imp,
    const int* __restrict__ start, float* __restrict__ pooled, int N, int G) {
    int g  = blockIdx.x;
    int t  = threadIdx.x;
    int tq = t & 31;   // float4 column: features 4*tq .. 4*tq+3
    int rs = t >> 5;   // row slot 0..3

    int s = start[g];
    int e = (g == G - 1) ? N : start[g + 1];
    int cnt = e - s;

    __shared__ float red[128];
    __shared__ float wbuf[128];
    __shared__ v4f   part[128];

    // pass 1: segment max of imp
    float m = -INFINITY;
    for (int i = s + t; i < e; i += 128) m = fmaxf(m, imp[i]);
    red[t] = m;
    __syncthreads();
    for (int off = 64; off > 0; off >>= 1) {
        if (t < off) red[t] = fmaxf(red[t], red[t + off]);
        __syncthreads();
    }
    float denom = red[0] * 10.0f;

    // pass 2: weighted sum, 128-node chunks, b128 loads
    v4f acc = {0.0f, 0.0f, 0.0f, 0.0f};
    for (int c = s; c < e; c += 128) {
        int chunk = min(128, e - c);
        __syncthreads();
        if (t < chunk) wbuf[t] = imp[c + t] / denom + 0.9f;
        __syncthreads();
        for (int i = rs; i < chunk; i += 4) {
            v4f xv = *(const v4f*)(x + (size_t)(c + i) * D + 4 * tq);
            float w = wbuf[i];
            acc += xv * w;
        }
    }

    // combine the 4 row-slot partials
    part[t] = acc;
    __syncthreads();
    if (rs == 0) {
        v4f tot = part[tq] + part[32 + tq] + part[64 + tq] + part[96 + tq];
        float div = (float)max(cnt, 1);
        tot /= div;
        *(v4f*)(pooled + (size_t)g * D + 4 * tq) = tot;
    }
}

// ---------------------------------------------------------------------------
// Kernel 3: fused MLP  out = relu(pooled@W1 + b1)@W2 + b2
// 256 threads = 8 waves per block. Block owns 16 rows of pooled; wave wv owns
// output columns [16*wv, 16*wv+16). Each GEMM = 32 x v_wmma_f32_16x16x4_f32.
//
// ISA fragment layouts (CDNA5 ISA):
//  A 16x4 f32 : lane l -> row m=l%16;  VGPR0=K(2*(l/16)),  VGPR1=K(2*(l/16)+1)
//  B 4x16 f32 : lane l -> col n=l%16;  VGPR0=K(2*(l/16)) row, VGPR1=next row
//  C/D 16x16  : VGPR r at lane l -> (m = r + 8*(l/16), n = l%16)
// ---------------------------------------------------------------------------
__global__ __launch_bounds__(256) void mlp_kernel(
    const float* __restrict__ pooled,
    const float* __restrict__ W1, const float* __restrict__ b1,
    const float* __restrict__ W2, const float* __restrict__ b2,
    float* __restrict__ out) {
    __shared__ float h_lds[16 * D];  // 8 KB relu(h) tile

    int wv   = threadIdx.x >> 5;   // wave id 0..7  -> column tile
    int lane = threadIdx.x & 31;
    int half = lane >> 4;          // 0: lanes 0-15, 1: lanes 16-31
    int n    = lane & 15;          // col within tile, also A-row within tile
    int rowbase = blockIdx.x * 16;
    int col0    = wv * 16;

    // ---- GEMM1: pooled @ W1 ----
    v8f c = {};
    #pragma unroll
    for (int kk = 0; kk < D; kk += 4) {
        int k0 = kk + 2 * half;
        v2f a;
        a.x = pooled[(size_t)(rowbase + n) * D + k0];
        a.y = pooled[(size_t)(rowbase + n) * D + k0 + 1];
        v2f b;
        b.x = W1[(size_t)k0 * D + col0 + n];
        b.y = W1[(size_t)(k0 + 1) * D + col0 + n];
        c = __builtin_amdgcn_wmma_f32_16x16x4_f32(
                false, a, false, b, (short)0, c, false, false);
    }

    // bias + relu -> LDS
    float bias1 = b1[col0 + n];
    #pragma unroll
    for (int r = 0; r < 8; ++r) {
        int ml = r + 8 * half;
        h_lds[ml * D + col0 + n] = fmaxf(c[r] + bias1, 0.0f);
    }
    __syncthreads();

    // ---- GEMM2: h @ W2 ----
    v8f c2 = {};
    #pragma unroll
    for (int kk = 0; kk < D; kk += 4) {
        int k0 = kk + 2 * half;
        v2f a;
        a.x = h_lds[n * D + k0];
        a.y = h_lds[n * D + k0 + 1];
        v2f b;
        b.x = W2[(size_t)k0 * D + col0 + n];
        b.y = W2[(size_t)(k0 + 1) * D + col0 + n];
        c2 = __builtin_amdgcn_wmma_f32_16x16x4_f32(
                false, a, false, b, (short)0, c2, false, false);
    }

    float bias2 = b2[col0 + n];
    #pragma unroll
    for (int r = 0; r < 8; ++r) {
        int ml = r + 8 * half;
        out[(size_t)(rowbase + ml) * D + col0 + n] = c2[r] + bias2;
    }
}

// ---------------------------------------------------------------------------
extern "C" void kernel_launch(void* const* d_in, const int* in_sizes, int n_in,
                              void* d_out, int out_size, void* d_ws, size_t ws_size,
                              hipStream_t stream) {
    const float* x     = (const float*)d_in[0];
    const float* imp   = (const float*)d_in[1];  // node_imp (N,1)
    const int*   batch = (const int*)d_in[2];
    const float* W1    = (const float*)d_in[3];
    const float* b1    = (const float*)d_in[4];
    const float* W2    = (const float*)d_in[5];
    const float* b2    = (const float*)d_in[6];
    float* out = (float*)d_out;

    int N = in_sizes[1];        // number of nodes
    int G = out_size / D;       // number of graphs (16384)

    // workspace layout: start[G] ints, then pooled[G*D] floats (256B aligned)
    char* wsb = (char*)d_ws;
    int*   start  = (int*)wsb;
    size_t poff   = (((size_t)G * sizeof(int)) + 255) & ~(size_t)255;
    float* pooled = (float*)(wsb + poff);

    seg_starts_kernel<<<(G + 255) / 256, 256, 0, stream>>>(batch, start, N, G);
    pool_kernel<<<G, 128, 0, stream>>>(x, imp, start, pooled, N, G);
    mlp_kernel<<<G / 16, 256, 0, stream>>>(pooled, W1, b1, W2, b2, out);
}